// RESCALScore_76124000354702
// MI455X (gfx1250) — compile-verified
//
#include <hip/hip_runtime.h>
#include <hip/hip_bf16.h>

typedef __attribute__((ext_vector_type(2))) float v2f;
typedef __attribute__((ext_vector_type(8))) float v8f;

#define ED 32    // entity dim
#define RD 32    // relation dim
#define EPB 128  // edges per block (8 wave32 tiles of 16)
#define TPB 256  // threads per block (8 waves)

// ---------------- bucketing kernels ----------------

__global__ void rescal_zero(int* __restrict__ counts, int n) {
    int i = blockIdx.x * blockDim.x + threadIdx.x;
    if (i < n) counts[i] = 0;
}

__global__ void rescal_hist(const int* __restrict__ rel_ids, int* __restrict__ counts, int E) {
    int i = blockIdx.x * blockDim.x + threadIdx.x;
    if (i < E) atomicAdd(&counts[rel_ids[i]], 1);
}

__global__ void rescal_scan(const int* __restrict__ counts,
                            int* __restrict__ offsets,
                            int* __restrict__ cursors,
                            int* __restrict__ blk_off, int n) {
    if (blockIdx.x == 0 && threadIdx.x == 0) {
        int run = 0, brun = 0;
        for (int r = 0; r < n; ++r) {
            offsets[r] = run;
            blk_off[r] = brun;
            cursors[r] = 0;
            run  += counts[r];
            brun += (counts[r] + EPB - 1) / EPB;
        }
        offsets[n] = run;
        blk_off[n] = brun;
    }
}

__global__ void rescal_scatter(const int* __restrict__ rel_ids,
                               const int* __restrict__ offsets,
                               int* __restrict__ cursors,
                               int* __restrict__ bucket, int E) {
    int i = blockIdx.x * blockDim.x + threadIdx.x;
    if (i < E) {
        int r = rel_ids[i];
        int pos = atomicAdd(&cursors[r], 1);
        bucket[offsets[r] + pos] = i;
    }
}

// ---------------- WMMA score kernel ----------------
// One 256-thread (8-wave) block per 128 edges of a single relation.
// M (32x32) staged to LDS once per block and shared by all 8 waves.
// Per wave: U = T_slab(16x32) x M^T via v_wmma_f32_16x16x4_f32 (8 K-steps,
// two accumulators for the two 16-col halves of U); score_e = dot(h_e, U_row).

__launch_bounds__(TPB)
__global__ void rescal_score_wmma(const float* __restrict__ node_emb,
                                  const float* __restrict__ rel_emb,
                                  const int* __restrict__ src,
                                  const int* __restrict__ dst,
                                  const int* __restrict__ offsets,  // n_rels+1
                                  const int* __restrict__ blk_off,  // n_rels+1
                                  const int* __restrict__ bucket,
                                  float* __restrict__ out,
                                  int n_rels) {
    __shared__ float M_lds[RD][ED + 1];     // padded stride 33 (bank spread)
    __shared__ float h_lds[EPB][ED + 1];
    __shared__ float t_lds[EPB][ED + 1];
    __shared__ int   eids[EPB];

    const int t = blockIdx.x;
    const int total = blk_off[n_rels];
    if (t >= total) return;                 // uniform per block

    // binary search: largest r with blk_off[r] <= t (uniform)
    int lo = 0, hi = n_rels;
    while (hi - lo > 1) {
        int mid = (lo + hi) >> 1;
        if (blk_off[mid] <= t) lo = mid; else hi = mid;
    }
    const int r     = lo;
    const int blk   = t - blk_off[r];
    const int base  = offsets[r];
    const int count = offsets[r + 1] - base;
    const int m0    = blk * EPB;
    const int nvb   = min(EPB, count - m0); // >= 1 for any existing block

    const int tid = threadIdx.x;

    // resolve edge ids (clamp padding slots to last valid edge)
    if (tid < EPB) {
        int m = tid < nvb ? tid : (nvb - 1);
        eids[tid] = bucket[base + m0 + m];
    }
    __syncthreads();

    // stage M: 1024 floats, 256 threads x 1 float4 (coalesced)
    {
        const float* Mg = rel_emb + (size_t)r * (RD * ED);
        int flat = tid * 4;
        float4 v = *(const float4*)(Mg + flat);
        int row = flat >> 5, col = flat & 31;
        M_lds[row][col + 0] = v.x; M_lds[row][col + 1] = v.y;
        M_lds[row][col + 2] = v.z; M_lds[row][col + 3] = v.w;
    }
    // stage head/tail slabs: 128 edges x 32 floats each (float4 per eighth-row)
#pragma unroll
    for (int j = 0; j < 4; ++j) {
        int flat = j * (TPB * 4) + tid * 4;
        int e = flat >> 5, col = flat & 31;
        int eid = eids[e];
        float4 hv = *(const float4*)(node_emb + (size_t)src[eid] * ED + col);
        float4 tv = *(const float4*)(node_emb + (size_t)dst[eid] * ED + col);
        h_lds[e][col + 0] = hv.x; h_lds[e][col + 1] = hv.y;
        h_lds[e][col + 2] = hv.z; h_lds[e][col + 3] = hv.w;
        t_lds[e][col + 0] = tv.x; t_lds[e][col + 1] = tv.y;
        t_lds[e][col + 2] = tv.z; t_lds[e][col + 3] = tv.w;
    }
    __syncthreads();

    // per-wave 16-edge tile
    const int lane  = tid & 31;
    const int w     = tid >> 5;             // wave id 0..7
    const int wbase = w * 16;
    const int wnv   = min(16, nvb - wbase); // wave-uniform; may be <= 0

    if (wnv > 0) {                          // wave-uniform branch: EXEC all-1s inside
        // fragment coords (ISA 7.12.2):
        //   A 16x4 f32: lane holds row m = lane&15, K = 2*(lane>>4) + {vgpr0,vgpr1}
        //   B 4x16 f32: lane holds col n = lane&15, K = 2*(lane>>4) + {vgpr0,vgpr1}
        //   C/D 16x16:  vgpr v, lane: row = v + 8*(lane>>4), col = lane&15
        const int c  = lane & 15;
        const int hs = lane >> 4;
        const int er = wbase + c;           // this lane's edge row in the slab

        v8f acc0 = {};                      // U columns 0..15
        v8f acc1 = {};                      // U columns 16..31
#pragma unroll
        for (int kk = 0; kk < 8; ++kk) {
            const int k0 = kk * 4 + 2 * hs;
            v2f a, b0, b1;
            a.x  = t_lds[er][k0];        a.y  = t_lds[er][k0 + 1];
            b0.x = M_lds[c][k0];         b0.y = M_lds[c][k0 + 1];      // B[k][n] = M[n][k]
            b1.x = M_lds[c + 16][k0];    b1.y = M_lds[c + 16][k0 + 1];
            acc0 = __builtin_amdgcn_wmma_f32_16x16x4_f32(false, a, false, b0,
                                                         (short)0, acc0, false, false);
            acc1 = __builtin_amdgcn_wmma_f32_16x16x4_f32(false, a, false, b1,
                                                         (short)0, acc1, false, false);
        }

        // partial dot with head: lane holds cols c and c+16 of rows v+8*hs
        float p[8];
#pragma unroll
        for (int v = 0; v < 8; ++v) {
            int row = v + 8 * hs;
            p[v] = acc0[v] * h_lds[wbase + row][c] + acc1[v] * h_lds[wbase + row][c + 16];
        }
        // reduce across the 16 lanes of each half (xor masks 1..8 stay within half)
#pragma unroll
        for (int v = 0; v < 8; ++v) {
            for (int off = 1; off < 16; off <<= 1)
                p[v] += __shfl_xor(p[v], off, 32);
        }
        // lane 0 -> rows 0..7, lane 16 -> rows 8..15
        if (c == 0) {
#pragma unroll
            for (int v = 0; v < 8; ++v) {
                int row = v + 8 * hs;
                if (row < wnv) out[eids[wbase + row]] = p[v];
            }
        }
    }
}

// ---------------- launch ----------------

extern "C" void kernel_launch(void* const* d_in, const int* in_sizes, int n_in,
                              void* d_out, int out_size, void* d_ws, size_t ws_size,
                              hipStream_t stream) {
    const float* node_emb = (const float*)d_in[0];
    const float* rel_emb  = (const float*)d_in[1];
    const int*   src      = (const int*)d_in[2];
    const int*   dst      = (const int*)d_in[3];
    const int*   rel_ids  = (const int*)d_in[4];
    float*       out      = (float*)d_out;

    const int E  = in_sizes[2];
    const int NR = in_sizes[1] / (RD * ED);

    int* ws      = (int*)d_ws;
    int* counts  = ws;             // NR
    int* offsets = ws + 256;       // NR+1
    int* cursors = ws + 512;       // NR
    int* blk_off = ws + 768;       // NR+1
    int* bucket  = ws + 1024;      // E

    rescal_zero<<<(NR + 255) / 256, 256, 0, stream>>>(counts, NR);
    rescal_hist<<<(E + 255) / 256, 256, 0, stream>>>(rel_ids, counts, E);
    rescal_scan<<<1, 1, 0, stream>>>(counts, offsets, cursors, blk_off, NR);
    rescal_scatter<<<(E + 255) / 256, 256, 0, stream>>>(rel_ids, offsets, cursors, bucket, E);

    const int maxBlocks = (E + EPB - 1) / EPB + NR;  // >= sum_r ceil(count_r/EPB)
    rescal_score_wmma<<<maxBlocks, TPB, 0, stream>>>(node_emb, rel_emb, src, dst,
                                                     offsets, blk_off, bucket, out, NR);
}